// SinkhornRouter_4492535791701
// MI455X (gfx1250) — compile-verified
//
#include <hip/hip_runtime.h>
#include <math.h>

typedef __attribute__((ext_vector_type(2))) float v2f;
typedef __attribute__((ext_vector_type(8))) float v8f;

#define EPS_F     1e-6f
#define NBATCH    4
#define NTOK      8192
#define NEXP      64
#define DIM       4096
#define MTOT      (NBATCH * NTOK)   /* 32768 flattened tokens */
#define NUM_ITERS 8
#define NSLICE    16                /* token slices for column reduction */

// ---------------------------------------------------------------------------
// GEMM (fp32 WMMA 16x16x4) + log(clip(.,eps)).  t0 written to d_out.
// Wave tile: 16 tokens x 64 experts. Block = 256 threads = 8 waves = 128 tokens.
// ---------------------------------------------------------------------------
__global__ __launch_bounds__(256) void sk_gemm_log_kernel(
    const float* __restrict__ x,   // [MTOT, DIM]
    const float* __restrict__ w,   // [NEXP, DIM]
    float* __restrict__ t0)        // [MTOT, NEXP]
{
    const int lane = threadIdx.x & 31;
    const int wave = threadIdx.x >> 5;
    const int row_base = blockIdx.x * 128 + wave * 16;
    const int half = lane >> 4;    // 0: K=k..k+1 ; 1: K=k+2..k+3
    const int m    = lane & 15;    // A: token row / B: expert row (within tile)

    const float* xr = x + (size_t)(row_base + m) * DIM + 2 * half;
    const float* wr = w + (size_t)m * DIM + 2 * half;

    v8f acc0 = {}, acc1 = {}, acc2 = {}, acc3 = {};

#pragma unroll 4
    for (int k = 0; k < DIM; k += 4) {
        v2f a  = *(const v2f*)(xr + k);
        v2f b0 = *(const v2f*)(wr + k);
        v2f b1 = *(const v2f*)(wr + 16 * DIM + k);
        v2f b2 = *(const v2f*)(wr + 32 * DIM + k);
        v2f b3 = *(const v2f*)(wr + 48 * DIM + k);
        acc0 = __builtin_amdgcn_wmma_f32_16x16x4_f32(false, a, false, b0, (short)0, acc0, false, false);
        acc1 = __builtin_amdgcn_wmma_f32_16x16x4_f32(false, a, false, b1, (short)0, acc1, false, false);
        acc2 = __builtin_amdgcn_wmma_f32_16x16x4_f32(false, a, false, b2, (short)0, acc2, false, false);
        acc3 = __builtin_amdgcn_wmma_f32_16x16x4_f32(false, a, false, b3, (short)0, acc3, false, false);
    }

    // C/D layout: VGPR v holds row M = v + 8*half, col N = lane&15.
#pragma unroll
    for (int v = 0; v < 8; ++v) {
        const int row = row_base + v + 8 * half;
        float* orow = t0 + (size_t)row * NEXP + m;
        orow[0]  = logf(fmaxf(acc0[v], EPS_F));
        orow[16] = logf(fmaxf(acc1[v], EPS_F));
        orow[32] = logf(fmaxf(acc2[v], EPS_F));
        orow[48] = logf(fmaxf(acc3[v], EPS_F));
    }
}

// ---------------------------------------------------------------------------
// Column pass stage 1: partial[b][slice][e] = sum_n exp(t0[b,n,e] - r[b,n])
// grid (NSLICE, NBATCH), block (64, 4). Coalesced over experts.
// ---------------------------------------------------------------------------
__global__ __launch_bounds__(256) void sk_col_partial_kernel(
    const float* __restrict__ t0,
    const float* __restrict__ r,
    float* __restrict__ partial,
    int use_r)
{
    const int e     = threadIdx.x;      // 0..63
    const int ty    = threadIdx.y;      // 0..3
    const int slice = blockIdx.x;       // 0..NSLICE-1
    const int b     = blockIdx.y;       // 0..NBATCH-1
    const int tok_per_slice = NTOK / NSLICE;   // 512
    const int n0 = slice * tok_per_slice;

    const float* tb = t0 + (size_t)b * NTOK * NEXP;
    const float* rb = r  + (size_t)b * NTOK;

    float s = 0.0f;
    for (int n = n0 + ty; n < n0 + tok_per_slice; n += 4) {
        float v = tb[(size_t)n * NEXP + e];
        if (use_r) v -= rb[n];
        s += expf(v);
    }

    __shared__ float sm[4][NEXP];
    sm[ty][e] = s;
    __syncthreads();
    if (ty == 0) {
        partial[((size_t)b * NSLICE + slice) * NEXP + e] =
            sm[0][e] + sm[1][e] + sm[2][e] + sm[3][e];
    }
}

// ---------------------------------------------------------------------------
// Column pass stage 2: c[b][e] = log(sum_slice partial).  1 block, 256 thr.
// ---------------------------------------------------------------------------
__global__ __launch_bounds__(256) void sk_col_combine_kernel(
    const float* __restrict__ partial,
    float* __restrict__ c)
{
    const int tid = threadIdx.x;
    const int b = tid >> 6, e = tid & 63;
    float s = 0.0f;
    for (int sl = 0; sl < NSLICE; ++sl)
        s += partial[((size_t)b * NSLICE + sl) * NEXP + e];
    c[b * NEXP + e] = logf(s);
}

// ---------------------------------------------------------------------------
// Row pass: r[b][n] = log(sum_e exp(t0[b,n,e] - c[b,e])).
// One wave32 per token; each lane handles 2 consecutive experts (b64 load).
// ---------------------------------------------------------------------------
__global__ __launch_bounds__(256) void sk_row_lse_kernel(
    const float* __restrict__ t0,
    const float* __restrict__ c,
    float* __restrict__ r)
{
    const int lane = threadIdx.x & 31;
    const int wave = threadIdx.x >> 5;
    const int gtok = blockIdx.x * 8 + wave;      // 0..MTOT-1
    const int b    = gtok >> 13;                 // /NTOK

    const float* row = t0 + (size_t)gtok * NEXP + 2 * lane;
    const float* cb  = c + b * NEXP + 2 * lane;

    v2f tv = *(const v2f*)row;
    float s = expf(tv[0] - cb[0]) + expf(tv[1] - cb[1]);

#pragma unroll
    for (int off = 16; off >= 1; off >>= 1)
        s += __shfl_xor(s, off, 32);

    if (lane == 0) r[gtok] = logf(s);
}

// ---------------------------------------------------------------------------
// Final: out = exp(t0 - c - r), in place on d_out.
// ---------------------------------------------------------------------------
__global__ __launch_bounds__(256) void sk_final_exp_kernel(
    float* __restrict__ t0,
    const float* __restrict__ c,
    const float* __restrict__ r)
{
    const size_t idx = (size_t)blockIdx.x * blockDim.x + threadIdx.x;
    const int    e   = (int)(idx & 63);
    const size_t tok = idx >> 6;
    const int    b   = (int)(tok >> 13);
    t0[idx] = expf(t0[idx] - c[b * NEXP + e] - r[tok]);
}

// ---------------------------------------------------------------------------
extern "C" void kernel_launch(void* const* d_in, const int* in_sizes, int n_in,
                              void* d_out, int out_size, void* d_ws, size_t ws_size,
                              hipStream_t stream)
{
    const float* x = (const float*)d_in[0];   // [4, 8192, 4096]
    const float* w = (const float*)d_in[1];   // [64, 4096]
    float* t0 = (float*)d_out;                // [4, 8192, 64] scratch -> final

    float* r       = (float*)d_ws;            // MTOT floats
    float* c       = r + MTOT;                // NBATCH*NEXP floats
    float* partial = c + NBATCH * NEXP;       // NBATCH*NSLICE*NEXP floats

    // 1) gates + log(clip) via fp32 WMMA
    sk_gemm_log_kernel<<<MTOT / 128, 256, 0, stream>>>(x, w, t0);

    // 2) 8 Sinkhorn iterations (collapsed to row/col offset updates)
    for (int it = 0; it < NUM_ITERS; ++it) {
        sk_col_partial_kernel<<<dim3(NSLICE, NBATCH), dim3(64, 4), 0, stream>>>(
            t0, r, partial, it > 0 ? 1 : 0);
        sk_col_combine_kernel<<<1, 256, 0, stream>>>(partial, c);
        sk_row_lse_kernel<<<MTOT / 8, 256, 0, stream>>>(t0, c, r);
    }

    // 3) out = exp(t0 - c - r)
    sk_final_exp_kernel<<<out_size / 256, 256, 0, stream>>>(t0, c, r);
}